// WFSA_40441412059662
// MI455X (gfx1250) — compile-verified
//
#include <hip/hip_runtime.h>

// ---------------------------------------------------------------------------
// WFSA scan on gfx1250: x_{t+1} = x_t^T * A[:, sym_t, :], T=4096 steps.
// Memory-bound (16 GB streamed, ~0.7 ms HBM floor; 64x4MB matrices mostly
// L2-resident). K-reduction done with V_WMMA_F32_16X16X4_F32 (f32 path keeps
// the 4096-step chain numerically sane). 64 waves x 16-column tiles cover the
// 1024 outputs; device-wide step barrier via atomic counter in d_ws.
// ---------------------------------------------------------------------------

typedef float v2f __attribute__((ext_vector_type(2)));
typedef float v8f __attribute__((ext_vector_type(8)));

#define QSTATES 1024
#define VSYM    64
#define TSTEPS  4096
#define NWG     16
#define TPB     128                 // 4 waves of 32
#define WAVES_PER_BLK 4
#define ROWSTRIDE (VSYM * QSTATES)  // stride (in floats) between q-rows for fixed symbol

// ---- init kernel: zero barrier counter, load x0 = init vector -------------
__global__ __launch_bounds__(QSTATES) void wfsa_init_kernel(
    const float* __restrict__ initv, float* __restrict__ xb0,
    unsigned* __restrict__ cnt)
{
    const int i = threadIdx.x;
    xb0[i] = initv[i];
    if (i == 0) *cnt = 0u;
}

// ---- main persistent scan kernel ------------------------------------------
__global__ __launch_bounds__(TPB, 1) void wfsa_scan_kernel(
    const float* __restrict__ A,       // [Q, V, Q]
    const int*   __restrict__ inputs,  // [T]
    const float* __restrict__ finalv,  // [Q]
    float*       __restrict__ out,     // [1]
    float*       __restrict__ xb0,     // ws: x buffer 0 (holds x_init at start)
    float*       __restrict__ xb1,     // ws: x buffer 1
    unsigned*    __restrict__ cnt)     // ws: barrier counter (zeroed by init)
{
    __shared__ float xs[QSTATES];

    const int tid   = threadIdx.x;
    const int lane  = tid & 31;
    const int wave  = tid >> 5;
    const int gwave = blockIdx.x * WAVES_PER_BLK + wave;   // 0..63
    const int col   = gwave * 16 + (lane & 15);            // output column
    const int k0    = (lane < 16) ? 0 : 2;                 // K sub-slot per ISA A/B layout

    float* bufs[2] = { xb0, xb1 };

    for (int t = 0; t < TSTEPS; ++t) {
        const float* __restrict__ xcur  = bufs[t & 1];
        float*       __restrict__ xnext = bufs[(t + 1) & 1];

        const int v = inputs[t];
        const float* __restrict__ Mv = A + (size_t)v * QSTATES;  // (q,r) at q*ROWSTRIDE + r

        // Stage x_t into LDS (4 KB) so the hot loop reads it via DS ops.
        for (int i = tid; i < QSTATES; i += TPB) xs[i] = xcur[i];
        __syncthreads();

        // Per-lane column pointers for B operand rows k0 and k0+1.
        const float* p0 = Mv + (size_t)k0 * ROWSTRIDE + col;
        const float* p1 = p0 + ROWSTRIDE;

        v8f c = {0.f, 0.f, 0.f, 0.f, 0.f, 0.f, 0.f, 0.f};

        // K-reduction over q in steps of 4 using V_WMMA_F32_16X16X4_F32.
        // A(16x4): every row = x[q..q+3]  -> lane supplies (x[q+k0], x[q+k0+1]).
        // B(4x16): rows = matrix rows q..q+3, cols = this wave's 16 outputs.
        // D row M=0 accumulates x_next for these 16 columns.
        #pragma unroll 4
        for (int q = 0; q < QSTATES; q += 4) {
            v2f a, b;
            const float2 xv = *(const float2*)&xs[q + k0];  // 8B aligned: q%4==0, k0 in {0,2}
            a.x = xv.x;  a.y = xv.y;
            b.x = *p0;   b.y = *p1;
            p0 += 4 * ROWSTRIDE;
            p1 += 4 * ROWSTRIDE;
            c = __builtin_amdgcn_wmma_f32_16x16x4_f32(
                    /*neg_a=*/false, a, /*neg_b=*/false, b,
                    /*c_mod=*/(short)0, c, /*reuse_a=*/false, /*reuse_b=*/false);
        }

        // D layout: VGPR0 of lanes 0..15 = (M=0, N=lane) -> x_next[col].
        if (lane < 16) xnext[col] = c[0];

        // Prefetch a slice of next step's matrix while we wait at the barrier
        // (symbol sequence is known ahead of time -> latency hiding).
        if (t + 1 < TSTEPS) {
            const int vn = inputs[t + 1];
            const float* pn = A + (size_t)vn * QSTATES
                                + (size_t)(lane * 32) * ROWSTRIDE + (size_t)(gwave * 16);
            __builtin_prefetch(pn, 0, 1);
        }

        // Device-wide step barrier: monotonic counter, target = NWG*(t+1).
        __threadfence();
        __syncthreads();
        if (tid == 0) {
            __hip_atomic_fetch_add(cnt, 1u, __ATOMIC_RELEASE, __HIP_MEMORY_SCOPE_AGENT);
            const unsigned target = (unsigned)NWG * (unsigned)(t + 1);
            while (__hip_atomic_load(cnt, __ATOMIC_ACQUIRE, __HIP_MEMORY_SCOPE_AGENT) < target) {
                __builtin_amdgcn_s_sleep(1);
            }
        }
        __syncthreads();
    }

    // Final dot with `final` (all-ones in the reference): block 0 reduces.
    if (blockIdx.x == 0) {
        __shared__ float red[TPB];
        const float* xfin = xb0;   // T even -> last write went to buffer 0
        float s = 0.f;
        for (int i = tid; i < QSTATES; i += TPB) s += xfin[i] * finalv[i];
        red[tid] = s;
        __syncthreads();
        for (int off = TPB / 2; off > 0; off >>= 1) {
            if (tid < off) red[tid] += red[tid + off];
            __syncthreads();
        }
        if (tid == 0) out[0] = red[0];
    }
}

// ---------------------------------------------------------------------------
extern "C" void kernel_launch(void* const* d_in, const int* in_sizes, int n_in,
                              void* d_out, int out_size, void* d_ws, size_t ws_size,
                              hipStream_t stream)
{
    (void)in_sizes; (void)n_in; (void)out_size; (void)ws_size;

    const float* A      = (const float*)d_in[0];  // (Q, V, Q) f32
    const float* initv  = (const float*)d_in[1];  // (Q,) f32
    const float* finalv = (const float*)d_in[2];  // (Q,) f32
    const int*   inputs = (const int*)d_in[3];    // (T,) i32
    float*       out    = (float*)d_out;

    float*    xb0 = (float*)d_ws;
    float*    xb1 = xb0 + QSTATES;
    unsigned* cnt = (unsigned*)(xb1 + QSTATES);

    wfsa_init_kernel<<<1, QSTATES, 0, stream>>>(initv, xb0, cnt);
    wfsa_scan_kernel<<<NWG, TPB, 0, stream>>>(A, inputs, finalv, out, xb0, xb1, cnt);
}